// MaxUnpooling2D_2199023256237
// MI455X (gfx1250) — compile-verified
//
#include <hip/hip_runtime.h>

// MaxUnpooling2D  (B,H,W,C)=(16,64,64,256), UP=(2,2)  ->  out (16,128,128,256) f32
//
// Memory-bound scatter re-expressed as a gather-style "expand": each thread owns
// one pooled cell's group of 4 consecutive channels, reads updates+mask once
// (B128, non-temporal), and writes the full 2x2 output window (4x B128 stores,
// non-temporal, fully coalesced).  Every output element is written exactly once,
// so no zero-init pass is needed.  Traffic = 402 MB total -> ~17 us at 23.3 TB/s.
// No matrix FLOPs exist in this op, so no WMMA is emitted by design.

typedef __attribute__((ext_vector_type(4))) float v4f;
typedef __attribute__((ext_vector_type(4))) int   v4i;

namespace {
constexpr int Bq = 16, Hq = 64, Wq = 64, Cq = 256;
constexpr int OH = Hq * 2;            // 128
constexpr int OW = Wq * 2;            // 128
constexpr int C4 = Cq / 4;            // 64 float4 channel-groups per (b,h,w)
constexpr int N4 = Bq * Hq * Wq * C4; // 4,194,304 threads (one float4 each)
constexpr int PFDIST = 4096;          // prefetch 4096 * 16B = 64 KB ahead
}

__global__ __launch_bounds__(256) void maxunpool2d_expand(
    const v4f* __restrict__ upd4,   // updates viewed as float4 [N4]
    const v4i* __restrict__ msk4,   // mask    viewed as int4   [N4]
    float* __restrict__ out)        // [B, 128, 128, 256]
{
    const int t = blockIdx.x * 256 + threadIdx.x;   // [0, N4)

    // Streaming prefetch of the two input streams (gfx1250 global_prefetch_b8).
    if (t + PFDIST < N4) {
        __builtin_prefetch((const void*)(upd4 + t + PFDIST), 0, 1);
        __builtin_prefetch((const void*)(msk4 + t + PFDIST), 0, 1);
    }

    // t -> (b, h, w, c4): channel-group fastest so wave32 lanes are contiguous
    // in channels => every load/store below is a coalesced 512B segment.
    const int c4  = t & (C4 - 1);
    const int bhw = t >> 6;
    const int w   = bhw & (Wq - 1);
    const int h   = (bhw >> 6) & (Hq - 1);
    const int b   = bhw >> 12;

    // Read-once inputs: non-temporal (don't pollute L2; output alone > L2).
    const v4f u = __builtin_nontemporal_load(upd4 + t);
    const v4i m = __builtin_nontemporal_load(msk4 + t);

    // mask = (y*OW + x)*C + f  with OW=128, C=256  ->  pure shifts/masks.
    const v4i ye = m >> 15;              // y in [0,128)
    const v4i xe = (m >> 8) & (OW - 1);  // x in [0,128)

    const int h2 = 2 * h;
    const int w2 = 2 * w;
    const unsigned obase =
        (((unsigned)b * OH + (unsigned)h2) * OW + (unsigned)w2) * Cq
        + (unsigned)c4 * 4;

    const v4i ubits = __builtin_bit_cast(v4i, u);

    // Write the full 2x2 window: selected position gets the update, the other
    // three get +0.0f (bitwise AND with the -1/0 compare mask; branchless).
#pragma unroll
    for (int wy = 0; wy < 2; ++wy) {
#pragma unroll
        for (int wx = 0; wx < 2; ++wx) {
            const v4i sel = (ye == (h2 + wy)) & (xe == (w2 + wx)); // -1 / 0 per lane-elem
            const v4f val = __builtin_bit_cast(v4f, ubits & sel);
            const unsigned off = obase + ((unsigned)wy * OW + (unsigned)wx) * Cq;
            __builtin_nontemporal_store(val, (v4f*)(out + off));
        }
    }
}

extern "C" void kernel_launch(void* const* d_in, const int* /*in_sizes*/, int /*n_in*/,
                              void* d_out, int /*out_size*/, void* /*d_ws*/, size_t /*ws_size*/,
                              hipStream_t stream) {
    const v4f* upd = (const v4f*)d_in[0];   // float32 updates [16,64,64,256]
    const v4i* msk = (const v4i*)d_in[1];   // int32  mask     [16,64,64,256]
    float*     out = (float*)d_out;         // float32 out     [16,128,128,256]

    static_assert(N4 % 256 == 0, "grid must tile exactly");
    dim3 grid(N4 / 256), block(256);
    maxunpool2d_expand<<<grid, block, 0, stream>>>(upd, msk, out);
}